// PropagatorModel_111669150399
// MI455X (gfx1250) — compile-verified
//
#include <hip/hip_runtime.h>

namespace {
constexpr int kH    = 512;
constexpr int kL    = 8;
constexpr int kK    = 128;
constexpr int kVD   = 256;
constexpr int kMLPH = 2048;   // MLP * H
constexpr int kT    = 512;
constexpr int kB    = 16;
constexpr int kVOC  = 4096;
constexpr int kNT   = 512;    // threads per pipeline block (16 waves)
constexpr int kNW   = 16;     // waves per pipeline block
constexpr int kRING = 8;      // ring-buffer depth between pipeline stages
constexpr float kWRITE   = 0.1f;
constexpr float kFORGET  = 0.02f;
constexpr float kKSCALE  = 0.08838834764831845f;  // 1/sqrt(128)
}

typedef __attribute__((ext_vector_type(2))) float v2f;
typedef __attribute__((ext_vector_type(8))) float v8f;

__device__ __forceinline__ v8f wmma_f32(v2f a, v2f b, v8f c) {
  // D = A(16x4 f32) * B(4x16 f32) + C(16x16 f32)
  return __builtin_amdgcn_wmma_f32_16x16x4_f32(false, a, false, b, (short)0, c,
                                               false, false);
}

__device__ __forceinline__ int ld_acq(int* p) {
  return __hip_atomic_load(p, __ATOMIC_ACQUIRE, __HIP_MEMORY_SCOPE_AGENT);
}
__device__ __forceinline__ void st_rel(int* p, int v) {
  __hip_atomic_store(p, v, __ATOMIC_RELEASE, __HIP_MEMORY_SCOPE_AGENT);
}

// C(16xN) = A(16xK, LDS row-major, stride lda) * W(KxN global, row stride ldw)
// If scaleCol != nullptr: C[r][c] += scaleCol[c] * acc   (accumulate mode)
// else:                   C[r][c]  = acc
// Waves split the N dimension into 16-wide tiles.
__device__ __forceinline__ void gemm16(const float* __restrict__ W, int ldw,
                                       int N, int K, const float* A, int lda,
                                       float* C, int ldc,
                                       const float* __restrict__ scaleCol,
                                       int wave, int lane) {
  const int lh = lane & 15;
  const int hi = (lane >> 4) & 1;
  for (int nb = wave * 16; nb < N; nb += kNW * 16) {
    const int col = nb + lh;
    v8f acc = {};
    for (int k0 = 0; k0 < K; k0 += 4) {
      const int ka = k0 + hi * 2;                 // lanes 16-31 hold K=2,3
      float2 av = *(const float2*)(A + lh * lda + ka);
      v2f a; a.x = av.x; a.y = av.y;
      v2f b; b.x = W[(size_t)ka * ldw + col];
      b.y = W[(size_t)(ka + 1) * ldw + col];
      acc = wmma_f32(a, b, acc);
    }
    if (scaleCol) {
      const float s = scaleCol[col];
#pragma unroll
      for (int r = 0; r < 8; ++r) C[(r + hi * 8) * ldc + col] += s * acc[r];
    } else {
#pragma unroll
      for (int r = 0; r < 8; ++r) C[(r + hi * 8) * ldc + col] = acc[r];
    }
  }
}

// Per-row rsqrt(mean(y^2)+eps); wave b reduces row b (16 rows == 16 waves).
__device__ __forceinline__ void row_rms(const float* Y, int D, float* rs,
                                        int wave, int lane) {
  float s = 0.f;
  for (int j = lane; j < D; j += 32) { float y = Y[wave * D + j]; s += y * y; }
#pragma unroll
  for (int off = 16; off > 0; off >>= 1) s += __shfl_xor(s, off, 32);
  if (lane == 0) rs[wave] = rsqrtf(s / (float)D + 1e-6f);
}

__global__ __launch_bounds__(kNT) void pipeline_kernel(
    const int* __restrict__ tok, const float* __restrict__ valid,
    const float* __restrict__ emb, const float* __restrict__ Wrk,
    const float* __restrict__ Wwk, const float* __restrict__ Wwv,
    const float* __restrict__ Wrp, const float* __restrict__ Wwg,
    const float* __restrict__ bwg, const float* __restrict__ Wfg,
    const float* __restrict__ bfg, const float* __restrict__ W1,
    const float* __restrict__ b1, const float* __restrict__ W2,
    const float* __restrict__ b2, const float* __restrict__ s1,
    const float* __restrict__ s2, const float* __restrict__ s3,
    const float* __restrict__ g1, const float* __restrict__ g2,
    const float* __restrict__ sfin, float* __restrict__ mstate,
    float* __restrict__ stage, float* __restrict__ xfin,
    int* __restrict__ flags) {
  alignas(16) __shared__ float xs[kB * kH];      // running activation x
  alignas(16) __shared__ float hs[kB * kH];      // rms'd activation
  alignas(16) __shared__ float as_[kB * kH];     // MLP hidden chunk
  alignas(16) __shared__ float rvb[kB * kVD];    // read value
  alignas(16) __shared__ float wvb[kB * kVD];    // write value -> err
  alignas(16) __shared__ float rkb[kB * kK];     // read key
  alignas(16) __shared__ float wkb[kB * kK];     // write key
  __shared__ float rsb[16];
  __shared__ float etas[16], fgs[16], validv[16];

  const int tid  = threadIdx.x;
  const int lane = tid & 31;
  const int wave = tid >> 5;
  const int l    = blockIdx.x;       // one workgroup per layer (pipeline stage)

  int* produced = flags;
  int* consumed = flags + kL;

  const float* Wrk_l = Wrk + (size_t)l * kH * kK;
  const float* Wwk_l = Wwk + (size_t)l * kH * kK;
  const float* Wwv_l = Wwv + (size_t)l * kH * kVD;
  const float* Wrp_l = Wrp + (size_t)l * kVD * kH;
  const float* Wwg_l = Wwg + (size_t)l * kH;
  const float* Wfg_l = Wfg + (size_t)l * kH;
  const float* W1_l  = W1  + (size_t)l * kH * kMLPH;
  const float* b1_l  = b1  + (size_t)l * kMLPH;
  const float* W2_l  = W2  + (size_t)l * kMLPH * kH;
  const float* b2_l  = b2  + (size_t)l * kH;
  const float* s1_l  = s1 + (size_t)l * kH;
  const float* s2_l  = s2 + (size_t)l * kH;
  const float* s3_l  = s3 + (size_t)l * kH;
  const float* g1_l  = g1 + (size_t)l * kH;
  const float* g2_l  = g2 + (size_t)l * kH;
  const float  bwg_l = bwg[l];
  const float  bfg_l = bfg[l];
  float* m_l = mstate + (size_t)l * kB * kK * kVD;

  for (int t = 0; t < kT; ++t) {
    // ---------------- acquire input activation ----------------
    if (l == 0) {
      for (int i = tid; i < kB * kH / 4; i += kNT) {
        int b = i >> 7, j4 = i & 127;                 // H/4 = 128
        ((float4*)xs)[i] =
            ((const float4*)(emb + (size_t)tok[t * kB + b] * kH))[j4];
      }
    } else {
      if (tid == 0) {
        while (ld_acq(&produced[l - 1]) < t + 1) __builtin_amdgcn_s_sleep(1);
      }
      __syncthreads();
      const float4* sg = (const float4*)(stage +
          ((size_t)(l - 1) * kRING + (t & (kRING - 1))) * kB * kH);
      for (int i = tid; i < kB * kH / 4; i += kNT) ((float4*)xs)[i] = sg[i];
      __syncthreads();
      if (tid == 0) st_rel(&consumed[l - 1], t + 1);
    }
    if (tid < kB) validv[tid] = valid[t * kB + tid];
    __syncthreads();

    // ---------------- h = rms(x) * s1 ----------------
    row_rms(xs, kH, rsb, wave, lane);
    __syncthreads();
    for (int i = tid; i < kB * kH / 4; i += kNT) {
      float r = rsb[i >> 7];
      float4 xv = ((const float4*)xs)[i];
      float4 sv = ((const float4*)s1_l)[i & 127];
      float4 h;
      h.x = xv.x * r * sv.x; h.y = xv.y * r * sv.y;
      h.z = xv.z * r * sv.z; h.w = xv.w * r * sv.w;
      ((float4*)hs)[i] = h;
    }
    __syncthreads();

    // ---------------- rk = rms(h @ Wrk) * kscale ----------------
    gemm16(Wrk_l, kK, kK, kH, hs, kH, rkb, kK, nullptr, wave, lane);
    __syncthreads();
    row_rms(rkb, kK, rsb, wave, lane);
    __syncthreads();
    for (int i = tid; i < kB * kK; i += kNT) rkb[i] *= rsb[i >> 7] * kKSCALE;
    __syncthreads();

    // ---------------- rv[b,v] = sum_k m[b,k,v] * rk[b,k] ----------------
    for (int idx = tid; idx < kB * (kVD / 4); idx += kNT) {
      int b = idx >> 6, v4 = (idx & 63) << 2;        // kVD/4 = 64
      const float4* mp = (const float4*)(m_l + (size_t)b * kK * kVD + v4);
      float4 acc = {0.f, 0.f, 0.f, 0.f};
      for (int k = 0; k < kK; ++k) {
        float4 mv = mp[k * (kVD / 4)];
        float rk = rkb[b * kK + k];
        acc.x += mv.x * rk; acc.y += mv.y * rk;
        acc.z += mv.z * rk; acc.w += mv.w * rk;
      }
      *(float4*)(rvb + b * kVD + v4) = acc;
    }
    __syncthreads();

    // ---------------- x += g1 * (rv @ Wrp) ----------------
    gemm16(Wrp_l, kH, kH, kVD, rvb, kVD, xs, kH, g1_l, wave, lane);
    __syncthreads();

    // ---------------- MLP: x += g2 * (silu(mi@W1+b1)@W2 + b2) -------------
    row_rms(xs, kH, rsb, wave, lane);
    __syncthreads();
    for (int i = tid; i < kB * kH / 4; i += kNT) {
      float r = rsb[i >> 7];
      float4 xv = ((const float4*)xs)[i];
      float4 sv = ((const float4*)s2_l)[i & 127];
      float4 h;
      h.x = xv.x * r * sv.x; h.y = xv.y * r * sv.y;
      h.z = xv.z * r * sv.z; h.w = xv.w * r * sv.w;
      ((float4*)hs)[i] = h;
    }
    __syncthreads();
    for (int c = 0; c < 4; ++c) {  // hidden split into 4 chunks of 512
      gemm16(W1_l + c * 512, kMLPH, 512, kH, hs, kH, as_, 512, nullptr, wave,
             lane);
      __syncthreads();
      for (int i = tid; i < kB * 512 / 4; i += kNT) {
        float4 a = ((const float4*)as_)[i];
        float4 bb = ((const float4*)(b1_l + c * 512))[i & 127];
        a.x += bb.x; a.y += bb.y; a.z += bb.z; a.w += bb.w;
        a.x = a.x / (1.f + expf(-a.x));
        a.y = a.y / (1.f + expf(-a.y));
        a.z = a.z / (1.f + expf(-a.z));
        a.w = a.w / (1.f + expf(-a.w));
        ((float4*)as_)[i] = a;
      }
      __syncthreads();
      gemm16(W2_l + (size_t)c * 512 * kH, kH, kH, 512, as_, 512, xs, kH, g2_l,
             wave, lane);
      __syncthreads();
    }
    for (int i = tid; i < kB * kH; i += kNT) {
      int j = i & (kH - 1);
      xs[i] += g2_l[j] * b2_l[j];
    }
    __syncthreads();

    // ---------------- publish x to next stage / final rms ----------------
    if (l < kL - 1) {
      if (tid == 0) {  // back-pressure: slot t%RING must be consumed
        int tgt = t + 1 - kRING;
        while (ld_acq(&consumed[l]) < tgt) __builtin_amdgcn_s_sleep(1);
      }
      __syncthreads();
      float4* sg = (float4*)(stage +
          ((size_t)l * kRING + (t & (kRING - 1))) * kB * kH);
      for (int i = tid; i < kB * kH / 4; i += kNT) sg[i] = ((const float4*)xs)[i];
      __threadfence();
      __syncthreads();
      if (tid == 0) st_rel(&produced[l], t + 1);
    } else {
      row_rms(xs, kH, rsb, wave, lane);
      __syncthreads();
      float4* xf = (float4*)(xfin + (size_t)t * kB * kH);
      for (int i = tid; i < kB * kH / 4; i += kNT) {
        float r = rsb[i >> 7];
        float4 xv = ((const float4*)xs)[i];
        float4 sv = ((const float4*)sfin)[i & 127];
        float4 o;
        o.x = xv.x * r * sv.x; o.y = xv.y * r * sv.y;
        o.z = xv.z * r * sv.z; o.w = xv.w * r * sv.w;
        xf[i] = o;
      }
      __syncthreads();  // before rsb reuse
    }

    // ---------------- write phase: w = rms(x) * s3 ----------------
    row_rms(xs, kH, rsb, wave, lane);
    __syncthreads();
    for (int i = tid; i < kB * kH / 4; i += kNT) {
      float r = rsb[i >> 7];
      float4 xv = ((const float4*)xs)[i];
      float4 sv = ((const float4*)s3_l)[i & 127];
      float4 h;
      h.x = xv.x * r * sv.x; h.y = xv.y * r * sv.y;
      h.z = xv.z * r * sv.z; h.w = xv.w * r * sv.w;
      ((float4*)hs)[i] = h;
    }
    __syncthreads();

    // wk = rms(w @ Wwk) * kscale
    gemm16(Wwk_l, kK, kK, kH, hs, kH, wkb, kK, nullptr, wave, lane);
    __syncthreads();
    row_rms(wkb, kK, rsb, wave, lane);
    __syncthreads();
    for (int i = tid; i < kB * kK; i += kNT) wkb[i] *= rsb[i >> 7] * kKSCALE;
    __syncthreads();

    // wv = tanh(w @ Wwv)
    gemm16(Wwv_l, kVD, kVD, kH, hs, kH, wvb, kVD, nullptr, wave, lane);
    __syncthreads();
    for (int i = tid; i < kB * kVD; i += kNT) wvb[i] = tanhf(wvb[i]);

    // gates: eta = sigmoid(w@Wwg + bwg)*WRITE, fg = sigmoid(w@Wfg + bfg)*FORGET
    {
      float se = 0.f, sf = 0.f;
      for (int j = lane; j < kH; j += 32) {
        float wv_ = hs[wave * kH + j];
        se += wv_ * Wwg_l[j];
        sf += wv_ * Wfg_l[j];
      }
#pragma unroll
      for (int off = 16; off > 0; off >>= 1) {
        se += __shfl_xor(se, off, 32);
        sf += __shfl_xor(sf, off, 32);
      }
      if (lane == 0) {
        etas[wave] = kWRITE / (1.f + expf(-(se + bwg_l)));
        fgs[wave]  = kFORGET / (1.f + expf(-(sf + bfg_l)));
      }
    }
    __syncthreads();

    // err = clip(wv - m^T wk, -1, 1)   (stored back into wvb)
    for (int idx = tid; idx < kB * (kVD / 4); idx += kNT) {
      int b = idx >> 6, v4 = (idx & 63) << 2;
      const float4* mp = (const float4*)(m_l + (size_t)b * kK * kVD + v4);
      float4 acc = {0.f, 0.f, 0.f, 0.f};
      for (int k = 0; k < kK; ++k) {
        float4 mv = mp[k * (kVD / 4)];
        float wk = wkb[b * kK + k];
        acc.x += mv.x * wk; acc.y += mv.y * wk;
        acc.z += mv.z * wk; acc.w += mv.w * wk;
      }
      float4 wv4 = *(const float4*)(wvb + b * kVD + v4);
      float4 e;
      e.x = fminf(fmaxf(wv4.x - acc.x, -1.f), 1.f);
      e.y = fminf(fmaxf(wv4.y - acc.y, -1.f), 1.f);
      e.z = fminf(fmaxf(wv4.z - acc.z, -1.f), 1.f);
      e.w = fminf(fmaxf(wv4.w - acc.w, -1.f), 1.f);
      *(float4*)(wvb + b * kVD + v4) = e;
    }
    __syncthreads();

    // m = vf*clip((1-fg)*m + eta*wk⊗err, -10, 10) + (1-vf)*m
    for (int idx = tid; idx < kB * kK * (kVD / 4); idx += kNT) {
      int b = idx >> 13;                 // kK*kVD/4 = 8192
      int r = idx & 8191;
      int k = r >> 6;                    // kVD/4 = 64
      int v4 = (r & 63) << 2;
      float4* mp = (float4*)(m_l + (size_t)b * kK * kVD + (size_t)k * kVD + v4);
      float4 mv = *mp;
      const float fa = 1.f - fgs[b];
      const float ew = etas[b] * wkb[b * kK + k];
      const float vf = validv[b];
      const float4 er = *(const float4*)(wvb + b * kVD + v4);
      float4 nm;
      nm.x = fminf(fmaxf(fa * mv.x + ew * er.x, -10.f), 10.f);
      nm.y = fminf(fmaxf(fa * mv.y + ew * er.y, -10.f), 10.f);
      nm.z = fminf(fmaxf(fa * mv.z + ew * er.z, -10.f), 10.f);
      nm.w = fminf(fmaxf(fa * mv.w + ew * er.w, -10.f), 10.f);
      mv.x = vf * nm.x + (1.f - vf) * mv.x;
      mv.y = vf * nm.y + (1.f - vf) * mv.y;
      mv.z = vf * nm.z + (1.f - vf) * mv.z;
      mv.w = vf * nm.w + (1.f - vf) * mv.w;
      *mp = mv;
    }
    __syncthreads();
  }
}

// logits[(t*B+b)][v] = xfin[t,b,:] . emb[v,:]   — dense f32 WMMA GEMM
// Each wave computes a 16x32 tile (two accumulators share one A fragment).
__global__ __launch_bounds__(256) void logits_kernel(
    const float* __restrict__ xfin, const float* __restrict__ emb,
    float* __restrict__ out) {
  const int lane = threadIdx.x & 31;
  const int wave = threadIdx.x >> 5;
  const int lh = lane & 15;
  const int hi = (lane >> 4) & 1;
  const int ntiles = kVOC / 32;                       // 128
  const int tile = blockIdx.x * 8 + wave;             // 8 waves per block
  const int mt = tile / ntiles;
  const int nt = tile - mt * ntiles;
  const int row0 = mt * 16;
  const int col0 = nt * 32;
  const float* Ab  = xfin + (size_t)(row0 + lh) * kH;      // A[m][k]
  const float* Bb0 = emb + (size_t)(col0 + lh) * kH;       // B[k][n] = emb[n][k]
  const float* Bb1 = emb + (size_t)(col0 + 16 + lh) * kH;
  v8f acc0 = {};
  v8f acc1 = {};
  for (int k0 = 0; k0 < kH; k0 += 4) {
    const int ka = k0 + hi * 2;
    float2 av = *(const float2*)(Ab + ka);
    float2 bv0 = *(const float2*)(Bb0 + ka);
    float2 bv1 = *(const float2*)(Bb1 + ka);
    v2f a; a.x = av.x; a.y = av.y;
    v2f b0; b0.x = bv0.x; b0.y = bv0.y;
    v2f b1; b1.x = bv1.x; b1.y = bv1.y;
    acc0 = wmma_f32(a, b0, acc0);
    acc1 = wmma_f32(a, b1, acc1);
  }
#pragma unroll
  for (int r = 0; r < 8; ++r) {
    out[(size_t)(row0 + r + hi * 8) * kVOC + col0 + lh] = acc0[r];
    out[(size_t)(row0 + r + hi * 8) * kVOC + col0 + 16 + lh] = acc1[r];
  }
}

extern "C" void kernel_launch(void* const* d_in, const int* in_sizes, int n_in,
                              void* d_out, int out_size, void* d_ws,
                              size_t ws_size, hipStream_t stream) {
  (void)in_sizes; (void)n_in; (void)out_size; (void)ws_size;
  const int*   tok   = (const int*)d_in[0];
  const float* valid = (const float*)d_in[1];
  const float* emb   = (const float*)d_in[2];
  const float* Wrk   = (const float*)d_in[3];
  const float* Wwk   = (const float*)d_in[4];
  const float* Wwv   = (const float*)d_in[5];
  const float* Wrp   = (const float*)d_in[6];
  const float* Wwg   = (const float*)d_in[7];
  const float* bwg   = (const float*)d_in[8];
  const float* Wfg   = (const float*)d_in[9];
  const float* bfg   = (const float*)d_in[10];
  const float* W1    = (const float*)d_in[11];
  const float* b1    = (const float*)d_in[12];
  const float* W2    = (const float*)d_in[13];
  const float* b2    = (const float*)d_in[14];
  const float* s1    = (const float*)d_in[15];
  const float* s2    = (const float*)d_in[16];
  const float* s3    = (const float*)d_in[17];
  const float* g1    = (const float*)d_in[18];
  const float* g2    = (const float*)d_in[19];
  const float* sfin  = (const float*)d_in[20];
  const float* initm = (const float*)d_in[21];

  char* ws = (char*)d_ws;
  int* flags = (int*)ws;                               // 256 B flag header
  float* wsf = (float*)(ws + 256);
  float* mstate = wsf;                                 // L*B*K*VD floats (16MB)
  float* xfin = mstate + (size_t)kL * kB * kK * kVD;   // T*B*H floats (16MB)
  float* stage = xfin + (size_t)kT * kB * kH;          // (L-1)*RING*B*H floats

  hipMemsetAsync(flags, 0, 256, stream);
  hipMemcpyAsync(mstate, initm, (size_t)kL * kB * kK * kVD * sizeof(float),
                 hipMemcpyDeviceToDevice, stream);

  pipeline_kernel<<<dim3(kL), dim3(kNT), 0, stream>>>(
      tok, valid, emb, Wrk, Wwk, Wwv, Wrp, Wwg, bwg, Wfg, bfg, W1, b1, W2, b2,
      s1, s2, s3, g1, g2, sfin, mstate, stage, xfin, flags);

  const int totalTiles = (kT * kB / 16) * (kVOC / 32);  // 512*128
  logits_kernel<<<dim3(totalTiles / 8), dim3(256), 0, stream>>>(
      xfin, emb, (float*)d_out);
}